// GraphEncoder_10402410791735
// MI455X (gfx1250) — compile-verified
//
#include <hip/hip_runtime.h>

typedef __attribute__((ext_vector_type(2))) float v2f;
typedef __attribute__((ext_vector_type(8))) float v8f;

// ---------------------------------------------------------------------------
// Zero a float buffer (grid-stride).
// ---------------------------------------------------------------------------
__global__ void zero_f32(float* __restrict__ p, long n) {
  long i = (long)blockIdx.x * blockDim.x + threadIdx.x;
  long stride = (long)gridDim.x * blockDim.x;
  for (; i < n; i += stride) p[i] = 0.0f;
}

// ---------------------------------------------------------------------------
// Degree count: cnt[dst[e]] += 1
// ---------------------------------------------------------------------------
__global__ void count_deg(const int* __restrict__ dst, float* __restrict__ cnt, int E) {
  int i = blockIdx.x * blockDim.x + threadIdx.x;
  if (i < E) atomicAdd(&cnt[dst[i]], 1.0f);
}

// ---------------------------------------------------------------------------
// C[M,32] = A[M,K] @ W[K,32] using V_WMMA_F32_16X16X4_F32 (fp32 WMMA).
// One wave per 16-row tile; the wave computes both 16-wide N tiles, reusing
// its A fragment. M must be a multiple of 16 (true here: 100000, 50000).
//
// Fragment layouts per CDNA5 ISA 7.12.2:
//   A 16x4 : VGPR j -> lanes 0-15: (M=lane, K=k+j), lanes 16-31: (M=lane-16, K=k+j+2)
//   B 4x16 : VGPR j -> lanes 0-15: (K=k+j, N=lane), lanes 16-31: (K=k+j+2, N=lane-16)
//   C 16x16: VGPR j -> lanes 0-15: (M=j,  N=lane), lanes 16-31: (M=j+8, N=lane-16)
// ---------------------------------------------------------------------------
__global__ __launch_bounds__(256) void gemm_n32(const float* __restrict__ A,
                                                const float* __restrict__ W,
                                                float* __restrict__ C,
                                                int M, int K) {
  int wid  = (blockIdx.x * blockDim.x + threadIdx.x) >> 5;  // global wave id = M tile
  int lane = threadIdx.x & 31;
  int m0 = wid * 16;
  if (m0 >= M) return;                       // whole-wave uniform exit; EXEC stays all-1 for WMMA

  int l15  = lane & 15;
  int koff = (lane >> 4) * 2;                // K sub-offset for this half-wave (0 or 2)
  const float* arow = A + (long)(m0 + l15) * K;

  v8f acc0 = {};                             // N = 0..15
  v8f acc1 = {};                             // N = 16..31
  for (int k = 0; k < K; k += 4) {
    v2f a = *(const v2f*)(arow + k + koff);  // A[m][k+koff], A[m][k+koff+1] (8B aligned)
    v2f b0, b1;
    b0.x = W[(k + koff + 0) * 32 + l15];
    b0.y = W[(k + koff + 1) * 32 + l15];
    b1.x = W[(k + koff + 0) * 32 + 16 + l15];
    b1.y = W[(k + koff + 1) * 32 + 16 + l15];
    acc0 = __builtin_amdgcn_wmma_f32_16x16x4_f32(false, a, false, b0, (short)0, acc0,
                                                 false, false);
    acc1 = __builtin_amdgcn_wmma_f32_16x16x4_f32(false, a, false, b1, (short)0, acc1,
                                                 false, false);
  }

  float* crow = C + (long)m0 * 32;
  int r = (lane >> 4) * 8;                   // C row offset for this half-wave
#pragma unroll
  for (int j = 0; j < 8; ++j) {
    crow[(j + r) * 32 + l15]      = acc0[j];
    crow[(j + r) * 32 + 16 + l15] = acc1[j];
  }
}

// ---------------------------------------------------------------------------
// Edge scatter of pre-projected 32-dim messages:
//   agg[dst[e]][f] += P[src[e]][f]
// One wave32 per edge, lane = feature -> 128B coalesced read + 128B
// contiguous f32 atomics (L2-resident accumulators).
// ---------------------------------------------------------------------------
__global__ __launch_bounds__(256) void scatter32(const float* __restrict__ P,
                                                 const int* __restrict__ src,
                                                 const int* __restrict__ dst,
                                                 float* __restrict__ agg, int E) {
  int wid  = (blockIdx.x * blockDim.x + threadIdx.x) >> 5;  // edge id
  int lane = threadIdx.x & 31;                              // feature id
  if (wid >= E) return;
  int s = src[wid];
  int d = dst[wid];
  float v = P[(long)s * 32 + lane];
  atomicAdd(&agg[(long)d * 32 + lane], v);
}

// ---------------------------------------------------------------------------
// out = relu(agg/max(cnt,1) + bias + self), thread per (node, feature).
// ---------------------------------------------------------------------------
__global__ void finalize32(const float* __restrict__ agg, const float* __restrict__ cnt,
                           const float* __restrict__ bias, const float* __restrict__ self,
                           float* __restrict__ out, long n) {
  long i = (long)blockIdx.x * blockDim.x + threadIdx.x;
  if (i < n * 32) {
    long node = i >> 5;
    int  f    = (int)(i & 31);
    float c = cnt[node];
    c = c > 1.0f ? c : 1.0f;
    float v = agg[i] / c + bias[f] + self[i];
    out[i] = v > 0.0f ? v : 0.0f;
  }
}

static inline int cdiv(long a, long b) { return (int)((a + b - 1) / b); }

extern "C" void kernel_launch(void* const* d_in, const int* in_sizes, int n_in,
                              void* d_out, int out_size, void* d_ws, size_t ws_size,
                              hipStream_t stream) {
  const float* x_user = (const float*)d_in[0];
  const float* x_item = (const float*)d_in[1];
  const int* e1_src = (const int*)d_in[2];
  const int* e1_dst = (const int*)d_in[3];
  const int* e2_src = (const int*)d_in[4];
  const int* e2_dst = (const int*)d_in[5];
  const float* Wl1_ui = (const float*)d_in[6];
  const float* bl1_ui = (const float*)d_in[7];
  const float* Wr1_ui = (const float*)d_in[8];
  const float* Wl1_iu = (const float*)d_in[9];
  const float* bl1_iu = (const float*)d_in[10];
  const float* Wr1_iu = (const float*)d_in[11];
  const float* Wl2_ui = (const float*)d_in[12];
  const float* bl2_ui = (const float*)d_in[13];
  const float* Wr2_ui = (const float*)d_in[14];
  const float* Wl2_iu = (const float*)d_in[15];
  const float* bl2_iu = (const float*)d_in[16];
  const float* Wr2_iu = (const float*)d_in[17];

  const long NU = in_sizes[0] / 128;   // 100000
  const long NI = in_sizes[1] / 64;    // 50000
  const int  E  = in_sizes[2];         // 1000000

  // Workspace layout (floats). agg_* / cnt_* kept contiguous for one-shot zeroing.
  float* w = (float*)d_ws;
  float* h_user   = w;                   w += NU * 32;
  float* h_item   = w;                   w += NI * 32;
  float* p_user   = w;                   w += NU * 32;   // projected messages
  float* p_item   = w;                   w += NI * 32;
  float* s_user   = w;                   w += NU * 32;   // self (lin_r) terms
  float* s_item   = w;                   w += NI * 32;
  float* agg_user = w;                   w += NU * 32;
  float* agg_item = w;                   w += NI * 32;
  float* cnt_user = w;                   w += NU;
  float* cnt_item = w;                   w += NI;

  const int TB = 256;
  const int WPB = TB / 32;  // waves per block

  // ---- zero aggregators + degree counters (one contiguous region) ----
  zero_f32<<<2048, TB, 0, stream>>>(agg_user, NU * 32 + NI * 32 + NU + NI);

  // ---- degree counts (shared by both layers: same edge lists) ----
  count_deg<<<cdiv(E, TB), TB, 0, stream>>>(e1_dst, cnt_item, E);
  count_deg<<<cdiv(E, TB), TB, 0, stream>>>(e2_dst, cnt_user, E);

  // ---- layer 1 projections (project-then-aggregate: mean(x)@Wl == mean(x@Wl)) ----
  gemm_n32<<<cdiv(NU / 16, WPB), TB, 0, stream>>>(x_user, Wl1_ui, p_user, (int)NU, 128);
  gemm_n32<<<cdiv(NI / 16, WPB), TB, 0, stream>>>(x_item, Wl1_iu, p_item, (int)NI, 64);
  gemm_n32<<<cdiv(NI / 16, WPB), TB, 0, stream>>>(x_item, Wr1_ui, s_item, (int)NI, 64);
  gemm_n32<<<cdiv(NU / 16, WPB), TB, 0, stream>>>(x_user, Wr1_iu, s_user, (int)NU, 128);

  // ---- layer 1 edge aggregation (32-dim messages) ----
  scatter32<<<cdiv(E, WPB), TB, 0, stream>>>(p_user, e1_src, e1_dst, agg_item, E);
  scatter32<<<cdiv(E, WPB), TB, 0, stream>>>(p_item, e2_src, e2_dst, agg_user, E);

  // ---- layer 1 combine + relu ----
  finalize32<<<cdiv(NI * 32, TB), TB, 0, stream>>>(agg_item, cnt_item, bl1_ui, s_item, h_item, NI);
  finalize32<<<cdiv(NU * 32, TB), TB, 0, stream>>>(agg_user, cnt_user, bl1_iu, s_user, h_user, NU);

  // ---- re-zero aggregators for layer 2 (counts reused) ----
  zero_f32<<<2048, TB, 0, stream>>>(agg_user, NU * 32 + NI * 32);

  // ---- layer 2 projections (K = 32) ----
  gemm_n32<<<cdiv(NU / 16, WPB), TB, 0, stream>>>(h_user, Wl2_ui, p_user, (int)NU, 32);
  gemm_n32<<<cdiv(NI / 16, WPB), TB, 0, stream>>>(h_item, Wl2_iu, p_item, (int)NI, 32);
  gemm_n32<<<cdiv(NI / 16, WPB), TB, 0, stream>>>(h_item, Wr2_ui, s_item, (int)NI, 32);
  gemm_n32<<<cdiv(NU / 16, WPB), TB, 0, stream>>>(h_user, Wr2_iu, s_user, (int)NU, 32);

  // ---- layer 2 edge aggregation ----
  scatter32<<<cdiv(E, WPB), TB, 0, stream>>>(p_user, e1_src, e1_dst, agg_item, E);
  scatter32<<<cdiv(E, WPB), TB, 0, stream>>>(p_item, e2_src, e2_dst, agg_user, E);

  // ---- layer 2 combine + relu -> d_out = [o_user (NU*32) | o_item (NI*32)] ----
  float* o_user = (float*)d_out;
  float* o_item = o_user + NU * 32;
  finalize32<<<cdiv(NI * 32, TB), TB, 0, stream>>>(agg_item, cnt_item, bl2_ui, s_item, o_item, NI);
  finalize32<<<cdiv(NU * 32, TB), TB, 0, stream>>>(agg_user, cnt_user, bl2_iu, s_user, o_user, NU);
}